// VectorQuantizerMultiHead_41205916238492
// MI455X (gfx1250) — compile-verified
//
#include <hip/hip_runtime.h>

typedef __attribute__((ext_vector_type(2))) float v2f;
typedef __attribute__((ext_vector_type(4))) float f4v;
typedef __attribute__((ext_vector_type(8))) float v8f;

#define B_TOT 32768
#define H_N   4
#define K_N   1024
#define D_N   1024
#define DH    256
#define EPAD  260   // 256 + 4 dword pad -> conflict-free ds_load_b64

// ---------------------------------------------------------------------------
// Kernel 1: n2half[h*K + k] = 0.5 * sum_d emb[h,k,d]^2   (one wave per row)
// ---------------------------------------------------------------------------
__global__ __launch_bounds__(256) void vq_n2_kernel(const float* __restrict__ emb,
                                                    float* __restrict__ n2half) {
    int wid  = (blockIdx.x * blockDim.x + threadIdx.x) >> 5;
    int lane = threadIdx.x & 31;
    if (wid >= H_N * K_N) return;
    const float* row = emb + (size_t)wid * DH;
    float s = 0.f;
#pragma unroll
    for (int c = 0; c < 2; ++c) {
        f4v v = *(const f4v*)(row + lane * 4 + c * 128);
        s += v.x * v.x + v.y * v.y + v.z * v.z + v.w * v.w;
    }
#pragma unroll
    for (int off = 16; off > 0; off >>= 1) s += __shfl_xor(s, off, 32);
    if (lane == 0) n2half[wid] = 0.5f * s;
}

// ---------------------------------------------------------------------------
// Kernel 2: WMMA score GEMM + argmax.
//   grid = (B/128, H), block = 256 (8 waves). Wave w owns rows
//   [b_base + 16w, b_base + 16w + 16). Loops over 64 N-tiles of 16 codes,
//   E tile staged in LDS (padded rows), A held in registers.
// ---------------------------------------------------------------------------
__global__ __launch_bounds__(256) void vq_score_kernel(const float* __restrict__ x,
                                                       const float* __restrict__ emb,
                                                       const float* __restrict__ n2half,
                                                       int*   __restrict__ codes_i,
                                                       float* __restrict__ codes_f) {
    __shared__ float lds_e[16 * EPAD];

    const int h      = blockIdx.y;
    const int b_base = blockIdx.x * 128;
    const int tid    = threadIdx.x;
    const int w      = tid >> 5;
    const int lane   = tid & 31;
    const int lr     = lane & 15;    // row-in-A-tile / col-in-B-tile
    const int hi     = lane >> 4;    // K-pair selector

    // --- preload A fragments for all 64 K-steps (16x4 fp32 layout) ---------
    const float* xrow = x + (size_t)(b_base + w * 16 + lr) * D_N + h * DH + 2 * hi;
    v2f a[64];
#pragma unroll
    for (int kk = 0; kk < 64; ++kk)
        a[kk] = *(const v2f*)(xrow + kk * 4);

    float best_v[8];
    int   best_i[8];
#pragma unroll
    for (int j = 0; j < 8; ++j) { best_v[j] = -3.0e38f; best_i[j] = 0; }

    const float* eh = emb + (size_t)h * K_N * DH;

    for (int nt = 0; nt < 64; ++nt) {
        // --- stage E tile (16 x 256 f32) into LDS, rows padded to EPAD -----
#pragma unroll
        for (int i = 0; i < 4; ++i) {
            int j  = i * 256 + tid;          // float4 index, 1024 total
            int r  = j >> 6;                 // 64 float4 per row
            int c4 = j & 63;
            f4v v = *(const f4v*)(eh + (size_t)(nt * 16 + r) * DH + c4 * 4);
            *(f4v*)(&lds_e[r * EPAD + c4 * 4]) = v;
        }
        __syncthreads();

        // prefetch next E tile toward L2 while we compute
        if (nt + 1 < 64) {
            const float* p = eh + (size_t)((nt + 1) * 16 + (tid >> 4)) * DH + (tid & 15) * 16;
            __builtin_prefetch(p, 0, 1);
        }

        // --- 64 WMMA K-steps: S(16x16) += A(16x4) * B(4x16) ----------------
        v8f acc = {};
#pragma unroll
        for (int kk = 0; kk < 64; ++kk) {
            v2f bfrag = *(const v2f*)(&lds_e[lr * EPAD + kk * 4 + 2 * hi]);
            acc = __builtin_amdgcn_wmma_f32_16x16x4_f32(
                false, a[kk], false, bfrag, (short)0, acc, false, false);
        }

        // --- running argmax of (dot - 0.5*||e||^2) --------------------------
        const int   kidx = nt * 16 + lr;
        const float hn   = n2half[h * K_N + kidx];
#pragma unroll
        for (int j = 0; j < 8; ++j) {
            float v = acc[j] - hn;
            if (v > best_v[j]) { best_v[j] = v; best_i[j] = kidx; }
        }
        __syncthreads();   // all waves done reading lds_e before restage
    }

    // --- reduce argmax across the 16 lanes of each half-wave ----------------
#pragma unroll
    for (int j = 0; j < 8; ++j) {
        float v   = best_v[j];
        int   idx = best_i[j];
#pragma unroll
        for (int off = 1; off < 16; off <<= 1) {
            float ov = __shfl_xor(v, off, 32);
            int   oi = __shfl_xor(idx, off, 32);
            if (ov > v || (ov == v && oi < idx)) { v = ov; idx = oi; }
        }
        best_v[j] = v;
        best_i[j] = idx;
    }

    // lanes 0 and 16 hold rows j and j+8 respectively
    if (lr == 0) {
#pragma unroll
        for (int j = 0; j < 8; ++j) {
            int row = b_base + w * 16 + j + 8 * hi;
            codes_i[row * H_N + h]          = best_i[j];
            codes_f[(size_t)row * H_N + h]  = (float)best_i[j];
        }
    }
}

// ---------------------------------------------------------------------------
// Kernel 3: gather quantized rows + loss. One block (4 waves = 4 heads) per b.
// ---------------------------------------------------------------------------
__global__ __launch_bounds__(128) void vq_gather_kernel(const float* __restrict__ x,
                                                        const float* __restrict__ emb,
                                                        const int* __restrict__ codes_i,
                                                        float* __restrict__ out_loss,
                                                        float* __restrict__ out_q) {
    __shared__ float part[4];
    int b    = blockIdx.x;
    int wv   = threadIdx.x >> 5;   // head
    int lane = threadIdx.x & 31;

    int code = codes_i[b * H_N + wv];
    const float* e  = emb + ((size_t)wv * K_N + code) * DH;
    const float* xr = x   + (size_t)b * D_N + wv * DH;
    float*       q  = out_q + (size_t)b * D_N + wv * DH;

    float s = 0.f;
#pragma unroll
    for (int c = 0; c < 2; ++c) {
        f4v ev = *(const f4v*)(e  + lane * 4 + c * 128);
        f4v xv = *(const f4v*)(xr + lane * 4 + c * 128);
        *(f4v*)(q + lane * 4 + c * 128) = ev;
        float dx = ev.x - xv.x, dy = ev.y - xv.y;
        float dz = ev.z - xv.z, dw = ev.w - xv.w;
        s += dx * dx + dy * dy + dz * dz + dw * dw;
    }
#pragma unroll
    for (int off = 16; off > 0; off >>= 1) s += __shfl_xor(s, off, 32);
    if (lane == 0) part[wv] = s;
    __syncthreads();
    if (threadIdx.x == 0) {
        float t = part[0] + part[1] + part[2] + part[3];
        out_loss[b] = 1.25f * t * (1.0f / 1024.0f);   // (1 + 0.25) * MSE over D
    }
}

// ---------------------------------------------------------------------------
extern "C" void kernel_launch(void* const* d_in, const int* in_sizes, int n_in,
                              void* d_out, int out_size, void* d_ws, size_t ws_size,
                              hipStream_t stream) {
    const float* x   = (const float*)d_in[0];   // (B, 1, D) f32
    const float* emb = (const float*)d_in[1];   // (H, K, dh) f32

    float* out       = (float*)d_out;
    float* out_loss  = out;                                    // B
    float* out_q     = out + B_TOT;                            // B*D
    float* out_codes = out + B_TOT + (size_t)B_TOT * D_N;      // B*H (as float)

    float* n2half  = (float*)d_ws;                             // 4096 f32
    int*   codes_i = (int*)((char*)d_ws + 16384);              // B*H int32

    // 4096 rows, 8 waves per 256-thread block -> 512 blocks
    vq_n2_kernel<<<512, 256, 0, stream>>>(emb, n2half);

    vq_score_kernel<<<dim3(B_TOT / 128, H_N), 256, 0, stream>>>(
        x, emb, n2half, codes_i, out_codes);

    vq_gather_kernel<<<B_TOT, 128, 0, stream>>>(x, emb, codes_i, out_loss, out_q);
}